// multiHeadAttention_54150947668413
// MI455X (gfx1250) — compile-verified
//
#include <hip/hip_runtime.h>

typedef __attribute__((ext_vector_type(2))) float v2f;
typedef __attribute__((ext_vector_type(8))) float v8f;
typedef __attribute__((ext_vector_type(4))) unsigned int u32x4;
typedef __attribute__((ext_vector_type(4))) int i32x4;
typedef __attribute__((ext_vector_type(8))) int i32x8;

namespace {
constexpr int Bb  = 32;
constexpr int Ss  = 22;
constexpr int Nn  = 2048;
constexpr int Cin = 2;
constexpr int Hh  = 8;
constexpr int Dk  = 8;
constexpr int Pp  = 6;
constexpr float NEGV = -32768.0f;          // -2^15, matches reference
constexpr int WAVES = 2;                   // waves per block
constexpr int NPW   = 8;                   // nodes per wave (sequential)
constexpr int NPBT  = WAVES * NPW;         // 16 nodes staged per block
constexpr int ROWS  = Hh * Ss;             // 176 packed (h,s) rows
constexpr int MT    = ROWS / 16;           // 11 M-tiles, zero padding
constexpr int ALD   = 24;                  // padded att row stride
constexpr int STG   = Ss * NPBT * Cin;     // 704 floats per staged tensor
}

#if __has_builtin(__builtin_amdgcn_tensor_load_to_lds)
#define HAVE_TDM 1
#else
#define HAVE_TDM 0
#endif

#if HAVE_TDM
// 2D TDM tile: `rows` rows of `rowElems` fp32, global row stride `strideElems`,
// packed contiguously into LDS at byte offset lds_off.
static __device__ inline void tdm_load_2d(unsigned lds_off, const float* gbase,
                                          int rowElems, int rows, int strideElems) {
  unsigned long long ga = (unsigned long long)(size_t)gbase;
  u32x4 g0;
  g0.x = 1u;                                            // count=1 (valid), user mode
  g0.y = lds_off;                                       // lds_addr (bytes)
  g0.z = (unsigned)(ga & 0xffffffffu);                  // global_addr[31:0]
  g0.w = (unsigned)((ga >> 32) & 0x01ffffffu)           // global_addr[56:32]
       | (2u << 30);                                    // type = 2 ("image")
  i32x8 g1;
  g1[0] = (2 << 16);                                    // data_size = 4B
  g1[1] = (rowElems & 0xffff) << 16;                    // tensor_dim0[15:0] @ bits63:48
  g1[2] = ((rowElems >> 16) & 0xffff)                   // tensor_dim0[31:16]
        | ((rows & 0xffff) << 16);                      // tensor_dim1[15:0]
  g1[3] = ((rows >> 16) & 0xffff)                       // tensor_dim1[31:16]
        | ((rowElems & 0xffff) << 16);                  // tile_dim0
  g1[4] = rows & 0xffff;                                // tile_dim1 (tile_dim2=0)
  g1[5] = strideElems;                                  // tensor_dim0_stride[31:0]
  g1[6] = 0;                                            // stride hi / dim1_stride lo
  g1[7] = 0;
  i32x4 z4 = {};
#if __clang_major__ >= 23
  i32x8 z8 = {};
  __builtin_amdgcn_tensor_load_to_lds(g0, g1, z4, z4, z8, 0);
#else
  __builtin_amdgcn_tensor_load_to_lds(g0, g1, z4, z4, 0);
#endif
}
#endif

__global__ __launch_bounds__(WAVES * 32)
void mha_wmma_kernel(
    const float* __restrict__ q, const float* __restrict__ k, const float* __restrict__ v,
    const unsigned char* __restrict__ mask,
    const float* __restrict__ Wq, const float* __restrict__ bq,
    const float* __restrict__ Wk, const float* __restrict__ bk,
    const float* __restrict__ Wv, const float* __restrict__ bv,
    const float* __restrict__ Wc, const float* __restrict__ bc,
    const float* __restrict__ Wf, const float* __restrict__ bf,
    float* __restrict__ out)
{
  __shared__ float sM[Hh][9];        // per-head 3x3 bilinear (incl. 1/DK^2)
  __shared__ float sAvc[Hh][3];      // folded V->Wc coefficients
  __shared__ unsigned int sMaskRow[Ss];
  __shared__ float sWf[Pp][Ss];
  __shared__ float sBf[Pp];
  __shared__ float sBc;
  __shared__ float sQ[STG];          // [Ss][NPBT][Cin], row-contiguous (TDM tile order)
  __shared__ float sK[STG];
  __shared__ float sV[STG];
  __shared__ float sAtt[WAVES][ROWS][ALD];
  __shared__ float sY[WAVES][Ss];

  const int tid  = threadIdx.x;
  const int wave = tid >> 5;
  const int lane = tid & 31;
  const int nt = Nn / NPBT;
  const int b  = blockIdx.x / nt;
  const int n0 = (blockIdx.x % nt) * NPBT;

#if HAVE_TDM
  // ---- async staging of q/k/v node tile via Tensor Data Mover ----
  if (wave == 0) {
    const size_t goff = ((size_t)b * Ss * Nn + n0) * Cin;
    tdm_load_2d((unsigned)(size_t)(void*)sQ, q + goff, NPBT * Cin, Ss, Nn * Cin);
    tdm_load_2d((unsigned)(size_t)(void*)sK, k + goff, NPBT * Cin, Ss, Nn * Cin);
    tdm_load_2d((unsigned)(size_t)(void*)sV, v + goff, NPBT * Cin, Ss, Nn * Cin);
  }
#endif

  // ---- prep (overlaps TDM): fold Wq/bq/Wk/bk into per-head 3x3 bilinear ----
  for (int idx = tid; idx < Hh * 9; idx += WAVES * 32) {
    int h = idx / 9, ij = idx % 9, i = ij / 3, jc = ij % 3;
    float acc = 0.f;
    for (int d = 0; d < Dk; ++d) {
      int cc = h * Dk + d;
      float aa = (i  < 2) ? Wq[cc * Cin + i ] : bq[cc];
      float bb = (jc < 2) ? Wk[cc * Cin + jc] : bk[cc];
      acc += aa * bb;
    }
    sM[h][ij] = acc * (1.0f / 64.0f);               // 1/DK^2 folded in
  }
  for (int idx = tid; idx < Hh * 3; idx += WAVES * 32) {
    int h = idx / 3, i = idx % 3;
    float acc = 0.f;
    for (int d = 0; d < Dk; ++d) {
      int cc = h * Dk + d;
      float aa = (i < 2) ? Wv[cc * Cin + i] : bv[cc];
      acc += aa * Wc[cc];
    }
    sAvc[h][i] = acc;
  }
  for (int s = tid; s < Ss; s += WAVES * 32) {
    unsigned bits = 0;
    for (int t = 0; t < Ss; ++t) bits |= (mask[s * Ss + t] ? 1u : 0u) << t;
    sMaskRow[s] = bits;
  }
  for (int idx = tid; idx < Pp * Ss; idx += WAVES * 32) sWf[idx / Ss][idx % Ss] = Wf[idx];
  if (tid < Pp) sBf[tid] = bf[tid];
  if (tid == 0) sBc = bc[0];

#if HAVE_TDM
  if (wave == 0) __builtin_amdgcn_s_wait_tensorcnt(0);
#else
  // fallback: synchronous cooperative staging (coalesced 128B rows)
  for (int idx = tid; idx < 3 * STG; idx += WAVES * 32) {
    int tsr = idx / STG, rem = idx % STG;
    int s  = rem / (NPBT * Cin);
    int rr = rem % (NPBT * Cin);
    const float* src = (tsr == 0) ? q : (tsr == 1) ? k : v;
    float* dst       = (tsr == 0) ? sQ : (tsr == 1) ? sK : sV;
    dst[rem] = src[((size_t)(b * Ss + s) * Nn + n0) * Cin + rr];
  }
#endif
  __syncthreads();

  const int  lm  = lane & 15;
  const bool hiL = lane >= 16;

  for (int nl = 0; nl < NPW; ++nl) {
    const int j = wave * NPW + nl;          // node slot in the staged tile

    // ---- B fragments (shared across heads): rows (k0, k1, 1, 0) ----
    // 32-bit B 4x16: VGPR0 = rows K0 (lanes0-15) / K2 (lanes16-31); VGPR1 = K1/K3
    v2f fB0, fB1;
    {
      float k0a = sK[(lm * NPBT + j) * Cin + 0];
      float k1a = sK[(lm * NPBT + j) * Cin + 1];
      fB0.x = hiL ? 1.0f : k0a;             // K2 row = ones
      fB0.y = hiL ? 0.0f : k1a;             // K3 row = zeros (K pad)
      int t1 = 16 + lm;
      float k0b = (t1 < Ss) ? sK[(t1 * NPBT + j) * Cin + 0] : 0.f;
      float k1b = (t1 < Ss) ? sK[(t1 * NPBT + j) * Cin + 1] : 0.f;
      fB1.x = hiL ? 1.0f : k0b;
      fB1.y = hiL ? 0.0f : k1b;
    }

    // ---- phase 1: logits via V_WMMA_F32_16X16X4_F32, rows packed r = h*22+s ----
#pragma unroll
    for (int m = 0; m < MT; ++m) {
      const int h0  = (m * 16) / Ss;        // compile-time
      const int thr = (h0 + 1) * Ss;        // compile-time
      int r = m * 16 + lm;
      int h = h0 + ((r >= thr) ? 1 : 0);    // branch-free div by 22
      int s = r - h * Ss;
      const float* Mh = sM[h];
      float q0 = sQ[(s * NPBT + j) * Cin + 0];
      float q1 = sQ[(s * NPBT + j) * Cin + 1];
      float axlo = fmaf(q0, Mh[0], fmaf(q1, Mh[3], Mh[6]));   // u[s][0]
      float aylo = fmaf(q0, Mh[1], fmaf(q1, Mh[4], Mh[7]));   // u[s][1]
      float axhi = fmaf(q0, Mh[2], fmaf(q1, Mh[5], Mh[8]));   // u[s][2]
      // 32-bit A 16x4: VGPR0 = K0 (lanes0-15) / K2 (lanes16-31); VGPR1 = K1/K3
      v2f fA;
      fA.x = hiL ? axhi : axlo;
      fA.y = hiL ? 0.0f : aylo;             // K3 pad
      v8f cz = {};
      v8f d0 = __builtin_amdgcn_wmma_f32_16x16x4_f32(false, fA, false, fB0, (short)0, cz, false, false);
      v8f d1 = __builtin_amdgcn_wmma_f32_16x16x4_f32(false, fA, false, fB1, (short)0, cz, false, false);
      // D layout: VGPR g holds M = g (lanes0-15) / g+8 (lanes16-31), N = lane%16
      int rowbase = m * 16 + (hiL ? 8 : 0);
#pragma unroll
      for (int g = 0; g < 8; ++g) {
        sAtt[wave][rowbase + g][lm]      = d0[g];
        sAtt[wave][rowbase + g][16 + lm] = d1[g];
      }
    }
    __syncthreads();

    // ---- phase 2: masked softmax + folded att@V@Wc, lane = query row s ----
    if (lane < Ss) {
      const int s = lane;
      const unsigned mrow = sMaskRow[s];
      float y = sBc;
      for (int h = 0; h < Hh; ++h) {
        const float* row = sAtt[wave][h * Ss + s];
        float lv[Ss];
        float mx = NEGV;
#pragma unroll
        for (int t = 0; t < Ss; ++t) {
          float x = ((mrow >> t) & 1u) ? row[t] : NEGV;
          lv[t] = x;
          mx = fmaxf(mx, x);
        }
        float a0 = sAvc[h][0], a1 = sAvc[h][1], a2 = sAvc[h][2];
        float sum = 0.f, acc = 0.f;
#pragma unroll
        for (int t = 0; t < Ss; ++t) {
          float e = __expf(lv[t] - mx);                 // v_exp_f32
          sum += e;
          float vc = fmaf(sV[(t * NPBT + j) * Cin + 0], a0,
                     fmaf(sV[(t * NPBT + j) * Cin + 1], a1, a2));
          acc = fmaf(e, vc, acc);
        }
        y += acc / sum;
      }
      sY[wave][s] = y;
    }
    __syncthreads();

    // ---- phase 3: Wf matvec, write out[b,p,n] ----
    if (lane < Pp) {
      float z = sBf[lane];
#pragma unroll
      for (int s = 0; s < Ss; ++s) z = fmaf(sWf[lane][s], sY[wave][s], z);
      out[((size_t)b * Pp + lane) * Nn + (n0 + j)] = z;
    }
  }
}

extern "C" void kernel_launch(void* const* d_in, const int* in_sizes, int n_in,
                              void* d_out, int out_size, void* d_ws, size_t ws_size,
                              hipStream_t stream) {
  const float* q  = (const float*)d_in[0];
  const float* k  = (const float*)d_in[1];
  const float* v  = (const float*)d_in[2];
  const unsigned char* mask = (const unsigned char*)d_in[3];
  const float* Wq = (const float*)d_in[4];
  const float* bq = (const float*)d_in[5];
  const float* Wk = (const float*)d_in[6];
  const float* bk = (const float*)d_in[7];
  const float* Wv = (const float*)d_in[8];
  const float* bv = (const float*)d_in[9];
  const float* Wc = (const float*)d_in[10];
  const float* bc = (const float*)d_in[11];
  const float* Wf = (const float*)d_in[12];
  const float* bf = (const float*)d_in[13];
  float* out = (float*)d_out;

  dim3 grid(Bb * (Nn / NPBT));    // 4096 blocks
  dim3 block(WAVES * 32);         // 2 waves, 8 nodes each
  hipLaunchKernelGGL(mha_wmma_kernel, grid, block, 0, stream,
                     q, k, v, mask, Wq, bq, Wk, bk, Wv, bv, Wc, bc, Wf, bf, out);
}